// DataEncoder_21423296872856
// MI455X (gfx1250) — compile-verified
//
#include <hip/hip_runtime.h>
#include <hip/hip_bf16.h>
#include <math.h>

typedef float v2f __attribute__((ext_vector_type(2)));
typedef float v8f __attribute__((ext_vector_type(8)));

#define LEAKY 0.2f
#define TOTALC 10000.0f
#define BNEPS 1e-5f
#define SOFTEPS 1e-7f

// ---------------------------------------------------------------------------
// l[row] = sum_k x[row,k]   (one block per row, coalesced strided reads)
// ---------------------------------------------------------------------------
__global__ void k_rowsum(const float* __restrict__ x, float* __restrict__ l, int F) {
    __shared__ float red[256];
    const int row = blockIdx.x;
    const float* xr = x + (size_t)row * F;
    float s = 0.f;
    for (int j = threadIdx.x; j < F; j += blockDim.x) s += xr[j];
    red[threadIdx.x] = s;
    __syncthreads();
    for (int off = 128; off > 0; off >>= 1) {
        if ((int)threadIdx.x < off) red[threadIdx.x] += red[threadIdx.x + off];
        __syncthreads();
    }
    if (threadIdx.x == 0) l[row] = red[0];
}

__global__ void k_zero(float* __restrict__ p, size_t n) {
    size_t i = (size_t)blockIdx.x * blockDim.x + threadIdx.x;
    if (i < n) p[i] = 0.f;
}

__global__ void k_deg(const int* __restrict__ dst, float* __restrict__ deg, int E) {
    int e = blockIdx.x * blockDim.x + threadIdx.x;
    if (e < E) unsafeAtomicAdd(&deg[dst[e]], 1.0f);
}

// dinv[i] = rsqrt(deg[i] + 1)   (in place: deg buffer becomes dinv)
__global__ void k_dinv(float* __restrict__ d, int n) {
    int i = blockIdx.x * blockDim.x + threadIdx.x;
    if (i < n) d[i] = rsqrtf(d[i] + 1.0f);
}

// ---------------------------------------------------------------------------
// Pack B[K x NB] into K-pair interleaved layout so one b64 load fetches a
// whole WMMA B-fragment:  Bp[((k/2)*NB + n)*2 + j] = B[k + j][n],  j in {0,1}
// ---------------------------------------------------------------------------
__global__ void k_pack(const float* __restrict__ B, float* __restrict__ Bp,
                       int K2, int NB) {
    int p = blockIdx.x * blockDim.x + threadIdx.x;
    if (p >= K2 * NB) return;
    int k2 = p / NB;
    int n  = p - k2 * NB;
    v2f v;
    v.x = B[(size_t)(2 * k2) * NB + n];
    v.y = B[(size_t)(2 * k2 + 1) * NB + n];
    ((v2f*)Bp)[p] = v;
}

// ---------------------------------------------------------------------------
// fp32 WMMA GEMM: C[M x NB] = A[M x KK] @ B[KK x NB]  (B pre-packed, see above)
// 256 threads = 8 waves; wave w -> rows [blockIdx.y*128 + 16w, +16),
// cols [blockIdx.x*16*NT, +16*NT). NB/KK compile-time -> immediate offsets.
// AMODE: 0 = raw A, 1 = A -> log1p(A * 10000 / lvec[row])
// EMODE: 0 = raw store, 1 = +bias[col], 2 = softplus(x+bias[col]) + eps
// ---------------------------------------------------------------------------
template <int NT, int NB, int KK, int AMODE, int EMODE>
__global__ void k_wmma_gemm(const float* __restrict__ A, const float* __restrict__ Bp,
                            float* __restrict__ C, const float* __restrict__ lvec,
                            const float* __restrict__ bias, int M) {
    const int wave  = threadIdx.x >> 5;
    const int lane  = threadIdx.x & 31;
    const int m0    = blockIdx.y * 128 + wave * 16;
    const int n0    = blockIdx.x * (16 * NT);
    const int nlane = lane & 15;
    const int mrow  = m0 + nlane;
    const int mc    = mrow < M ? mrow : (M - 1);     // clamp loads
    const int kh    = (lane >> 4) << 1;              // 0 or 2

    float scale = 0.f;
    if (AMODE == 1) scale = TOTALC / lvec[mc];

    v8f acc[NT];
#pragma unroll
    for (int t = 0; t < NT; ++t) acc[t] = {};

    const float* arow  = A + (size_t)mc * KK + kh;                    // 8B-aligned
    const float* bbase = Bp + ((size_t)(kh >> 1) * NB + n0 + nlane) * 2;

    for (int k = 0; k < KK; k += 4) {
        v2f a = *(const v2f*)(arow + k);
        if (AMODE == 1) {
            a.x = log1pf(a.x * scale);
            a.y = log1pf(a.y * scale);
        }
        const float* bp = bbase + (size_t)k * NB;    // (k/2)*NB pairs * 2 floats
        v2f bfrag[NT];
#pragma unroll
        for (int t = 0; t < NT; ++t)                 // batch all B loads (b64 each)
            bfrag[t] = *(const v2f*)(bp + t * 32);
#pragma unroll
        for (int t = 0; t < NT; ++t)                 // independent WMMAs pipeline
            acc[t] = __builtin_amdgcn_wmma_f32_16x16x4_f32(false, a, false, bfrag[t],
                                                           (short)0, acc[t],
                                                           false, false);
    }

    // epilogue: m0 is wave-uniform -> force the edge predicate into an SGPR so
    // the full/partial selection is a scalar branch (no exec-mask ladders)
    const int rbase = m0 + ((lane >> 4) << 3);
    const int fullu = __builtin_amdgcn_readfirstlane((m0 + 16) <= M ? 1 : 0);
#pragma unroll
    for (int t = 0; t < NT; ++t) {
        const int col = n0 + t * 16 + nlane;
        float bv = 0.f;
        if (EMODE >= 1) bv = bias[col];
        float* cp = C + (size_t)rbase * NB + col;
        if (fullu) {
#pragma unroll
            for (int r = 0; r < 8; ++r) {
                float v = acc[t][r];
                if (EMODE == 1) v += bv;
                if (EMODE == 2) {
                    v += bv;
                    v = (v > 20.f) ? v : log1pf(expf(v));
                    v += SOFTEPS;
                }
                cp[r * NB] = v;                      // immediate offsets (NB const)
            }
        } else {
#pragma unroll
            for (int r = 0; r < 8; ++r) {
                if (rbase + r < M) {
                    float v = acc[t][r];
                    if (EMODE == 1) v += bv;
                    if (EMODE == 2) {
                        v += bv;
                        v = (v > 20.f) ? v : log1pf(expf(v));
                        v += SOFTEPS;
                    }
                    cp[r * NB] = v;
                }
            }
        }
    }
}

// Naive fallback GEMM for non-reference shapes (not taken for the given inputs)
template <int AMODE, int EMODE>
__global__ void k_gemm_naive(const float* __restrict__ A, const float* __restrict__ B,
                             float* __restrict__ C, const float* __restrict__ lvec,
                             const float* __restrict__ bias, int M, int Nb, int K) {
    size_t i = (size_t)blockIdx.x * blockDim.x + threadIdx.x;
    if (i >= (size_t)M * Nb) return;
    int row = (int)(i / Nb), col = (int)(i - (size_t)row * Nb);
    float scale = (AMODE == 1) ? TOTALC / lvec[row] : 0.f;
    float s = 0.f;
    for (int k = 0; k < K; ++k) {
        float a = A[(size_t)row * K + k];
        if (AMODE == 1) a = log1pf(a * scale);
        s += a * B[(size_t)k * Nb + col];
    }
    if (EMODE == 1) s += bias[col];
    if (EMODE == 2) {
        s += bias[col];
        s = (s > 20.f) ? s : log1pf(expf(s));
        s += SOFTEPS;
    }
    C[i] = s;
}

// hout = dinv[row]^2 * hin    (self-loop term; also initializes the agg buffer)
__global__ void k_selfinit(const float* __restrict__ hin, float* __restrict__ hout,
                           const float* __restrict__ dinv, size_t total, int hshift) {
    size_t i = (size_t)blockIdx.x * blockDim.x + threadIdx.x;
    if (i >= total) return;
    int row = (int)(i >> hshift);
    float di = dinv[row];
    hout[i] = di * di * hin[i];
}

// hout[dst,:] += dinv[src]*dinv[dst] * hin[src,:]
// H/4 lanes per edge, float4 gather, hardware fp32 atomics (no CAS)
__global__ void k_agg(const float* __restrict__ hin, float* __restrict__ hout,
                      const int* __restrict__ src, const int* __restrict__ dst,
                      const float* __restrict__ dinv, size_t total, int H,
                      int chshift) {
    size_t tid = (size_t)blockIdx.x * blockDim.x + threadIdx.x;
    if (tid >= total) return;
    int e  = (int)(tid >> chshift);
    int ch = (int)(tid & ((1u << chshift) - 1));
    int s = src[e], d = dst[e];
    float norm = dinv[s] * dinv[d];
    float4 v = ((const float4*)(hin + (size_t)s * H))[ch];
    float* o = hout + (size_t)d * H + (size_t)ch * 4;
    unsafeAtomicAdd(o + 0, v.x * norm);
    unsafeAtomicAdd(o + 1, v.y * norm);
    unsafeAtomicAdd(o + 2, v.z * norm);
    unsafeAtomicAdd(o + 3, v.w * norm);
}

// per-column sum / sumsq partials (thread = column -> fully coalesced rows)
__global__ void k_bnstats(const float* __restrict__ h, float* __restrict__ sum,
                          float* __restrict__ sumsq, int Nn, int H) {
    const int col = threadIdx.x;
    int r0 = blockIdx.x * 64;
    int r1 = r0 + 64 < Nn ? r0 + 64 : Nn;
    float s = 0.f, s2 = 0.f;
    for (int r = r0; r < r1; ++r) {
        float v = h[(size_t)r * H + col];
        s += v;
        s2 += v * v;
    }
    unsafeAtomicAdd(&sum[col], s);
    unsafeAtomicAdd(&sumsq[col], s2);
}

__global__ void k_bnfinal(const float* __restrict__ sum, const float* __restrict__ sumsq,
                          float* __restrict__ mu, float* __restrict__ rstd, int Nn) {
    int c = blockIdx.x * blockDim.x + threadIdx.x;
    float m = sum[c] / (float)Nn;
    float var = sumsq[c] / (float)Nn - m * m;   // biased var (matches jnp.var)
    mu[c] = m;
    rstd[c] = rsqrtf(var + BNEPS);
}

// h = LeakyReLU( gamma*(h-mu)*rstd + beta )   in place
__global__ void k_bnact(float* __restrict__ h, const float* __restrict__ gamma,
                        const float* __restrict__ beta, const float* __restrict__ mu,
                        const float* __restrict__ rstd, size_t total, int hmask) {
    size_t i = (size_t)blockIdx.x * blockDim.x + threadIdx.x;
    if (i >= total) return;
    int c = (int)(i & (size_t)hmask);
    float v = gamma[c] * (h[i] - mu[c]) * rstd[c] + beta[c];
    h[i] = v >= 0.f ? v : LEAKY * v;
}

// ---------------------------------------------------------------------------
extern "C" void kernel_launch(void* const* d_in, const int* in_sizes, int n_in,
                              void* d_out, int out_size, void* d_ws, size_t ws_size,
                              hipStream_t stream) {
    const float* x   = (const float*)d_in[0];
    const float* W1  = (const float*)d_in[1];
    const float* g1  = (const float*)d_in[2];
    const float* be1 = (const float*)d_in[3];
    const float* W2  = (const float*)d_in[4];
    const float* g2  = (const float*)d_in[5];
    const float* be2 = (const float*)d_in[6];
    const float* Wl  = (const float*)d_in[7];
    const float* bl  = (const float*)d_in[8];
    const float* Ws  = (const float*)d_in[9];
    const float* bs  = (const float*)d_in[10];
    const int*   ei  = (const int*)d_in[11];

    const int H    = in_sizes[2];            // 256
    const int Fin  = in_sizes[1] / H;        // 1000
    const int Nn   = in_sizes[0] / Fin;      // 50000
    const int Fout = in_sizes[8];            // 32
    const int E    = in_sizes[11] / 2;       // 800000
    const int* src = ei;
    const int* dst = ei + E;

    const int hshift  = __builtin_ctz(H);        // 8
    const int chshift = __builtin_ctz(H >> 2);   // 6

    float* out     = (float*)d_out;
    float* loc_out = out;
    float* std_out = out + (size_t)Nn * Fout;
    float* l_out   = out + (size_t)2 * Nn * Fout;

    // workspace layout (floats)
    float* ws = (float*)d_ws;
    size_t o = 0;
    float* hA     = ws + o; o += (size_t)Nn * H;
    float* hB     = ws + o; o += (size_t)Nn * H;
    float* dinv   = ws + o; o += (size_t)Nn + (Nn & 1);   // keep 8B alignment
    float* bnsum  = ws + o; o += (size_t)H;
    float* bnss   = ws + o; o += (size_t)H;
    float* bnmu   = ws + o; o += (size_t)H;
    float* bnrstd = ws + o; o += (size_t)H;
    float* W1p    = ws + o; o += (size_t)Fin * H;
    float* W2p    = ws + o; o += (size_t)H * H;
    float* Wlp    = ws + o; o += (size_t)H * Fout;
    float* Wsp    = ws + o; o += (size_t)H * Fout;

    const size_t NH = (size_t)Nn * H;
    auto blks = [](size_t n, int t) { return (unsigned)((n + t - 1) / t); };

    // 1) library size l (output #3)
    k_rowsum<<<Nn, 256, 0, stream>>>(x, l_out, Fin);

    // 2) degree -> dinv
    k_zero<<<blks(Nn, 256), 256, 0, stream>>>(dinv, (size_t)Nn);
    k_deg<<<blks(E, 256), 256, 0, stream>>>(dst, dinv, E);
    k_dinv<<<blks(Nn, 256), 256, 0, stream>>>(dinv, Nn);

    const bool fast = (H == 256) && (Fin == 1000) && (Fout == 32);
    const int MB = (Nn + 127) / 128;
    dim3 gmain(2, MB);                            // 256 / (16*8)
    dim3 ghead(1, MB);                            // 32 / (16*2)
    const size_t aggTotal = (size_t)E << chshift;

    if (fast) {
        // pre-pack weights into b64-fragment layout (one-off, ~1.3 MB total)
        k_pack<<<blks((size_t)Fin / 2 * H, 256), 256, 0, stream>>>(W1, W1p, Fin / 2, H);
        k_pack<<<blks((size_t)H / 2 * H, 256), 256, 0, stream>>>(W2, W2p, H / 2, H);
        k_pack<<<blks((size_t)H / 2 * Fout, 256), 256, 0, stream>>>(Wl, Wlp, H / 2, Fout);
        k_pack<<<blks((size_t)H / 2 * Fout, 256), 256, 0, stream>>>(Ws, Wsp, H / 2, Fout);
    }

    // 3) layer 1: hA = log1p(x*10000/l) @ W1   (normalize fused into A-load)
    if (fast)
        k_wmma_gemm<8, 256, 1000, 1, 0><<<gmain, 256, 0, stream>>>(x, W1p, hA, l_out,
                                                                   nullptr, Nn);
    else
        k_gemm_naive<1, 0><<<blks(NH, 256), 256, 0, stream>>>(x, W1, hA, l_out,
                                                              nullptr, Nn, H, Fin);
    // GCN aggregate: hB = dinv^2*hA + scatter(norm*hA[src] -> dst)
    k_selfinit<<<blks(NH, 256), 256, 0, stream>>>(hA, hB, dinv, NH, hshift);
    k_agg<<<blks(aggTotal, 256), 256, 0, stream>>>(hA, hB, src, dst, dinv,
                                                   aggTotal, H, chshift);
    // BN + LeakyReLU
    k_zero<<<blks(2 * H, 256), 256, 0, stream>>>(bnsum, (size_t)2 * H);
    k_bnstats<<<blks(Nn, 64), H, 0, stream>>>(hB, bnsum, bnss, Nn, H);
    k_bnfinal<<<1, H, 0, stream>>>(bnsum, bnss, bnmu, bnrstd, Nn);
    k_bnact<<<blks(NH, 256), 256, 0, stream>>>(hB, g1, be1, bnmu, bnrstd, NH, H - 1);

    // 4) layer 2: hA = hB @ W2 ; aggregate into hB ; BN + act
    if (fast)
        k_wmma_gemm<8, 256, 256, 0, 0><<<gmain, 256, 0, stream>>>(hB, W2p, hA,
                                                                  nullptr, nullptr, Nn);
    else
        k_gemm_naive<0, 0><<<blks(NH, 256), 256, 0, stream>>>(hB, W2, hA, nullptr,
                                                              nullptr, Nn, H, H);
    k_selfinit<<<blks(NH, 256), 256, 0, stream>>>(hA, hB, dinv, NH, hshift);
    k_agg<<<blks(aggTotal, 256), 256, 0, stream>>>(hA, hB, src, dst, dinv,
                                                   aggTotal, H, chshift);
    k_zero<<<blks(2 * H, 256), 256, 0, stream>>>(bnsum, (size_t)2 * H);
    k_bnstats<<<blks(Nn, 64), H, 0, stream>>>(hB, bnsum, bnss, Nn, H);
    k_bnfinal<<<1, H, 0, stream>>>(bnsum, bnss, bnmu, bnrstd, Nn);
    k_bnact<<<blks(NH, 256), 256, 0, stream>>>(hB, g2, be2, bnmu, bnrstd, NH, H - 1);

    // 5) heads: bias / softplus epilogues fused into the WMMA store
    if (fast) {
        k_wmma_gemm<2, 32, 256, 0, 1><<<ghead, 256, 0, stream>>>(hB, Wlp, loc_out,
                                                                 nullptr, bl, Nn);
        k_wmma_gemm<2, 32, 256, 0, 2><<<ghead, 256, 0, stream>>>(hB, Wsp, std_out,
                                                                 nullptr, bs, Nn);
    } else {
        const size_t NF = (size_t)Nn * Fout;
        k_gemm_naive<0, 1><<<blks(NF, 256), 256, 0, stream>>>(hB, Wl, loc_out,
                                                              nullptr, bl, Nn, Fout, H);
        k_gemm_naive<0, 2><<<blks(NF, 256), 256, 0, stream>>>(hB, Ws, std_out,
                                                              nullptr, bs, Nn, Fout, H);
    }
}